// HiBERT_65403761983936
// MI455X (gfx1250) — compile-verified
//
#include <hip/hip_runtime.h>
#include <hip/hip_bf16.h>

// Problem constants (match reference)
#define BB 32
#define TT 4096
#define HH 768
#define NN 128
#define CC 16

#define BDIM 256   // 8 waves of 32
#define NWAVE 8
#define HTILES (HH / 16)   // 48

typedef __attribute__((ext_vector_type(2))) float v2f;
typedef __attribute__((ext_vector_type(8))) float v8f;

__device__ __forceinline__ float wave_reduce_add(float v) {
    // wave32 butterfly reduction
    v += __shfl_xor(v, 16, 32);
    v += __shfl_xor(v, 8, 32);
    v += __shfl_xor(v, 4, 32);
    v += __shfl_xor(v, 2, 32);
    v += __shfl_xor(v, 1, 32);
    return v;
}

__global__ __launch_bounds__(BDIM) void hibert_kernel(
    const float* __restrict__ hidden,      // [B,T,H]
    const float* __restrict__ W,           // [1,2H]
    const float* __restrict__ bias,        // [1]
    const int*   __restrict__ cls_pos,     // [B,N]
    const int*   __restrict__ chunk_ids,   // [B,N]
    float*       __restrict__ out)         // [B,N]
{
    __shared__ float s_sums[CC][HH];   // 49152 B : per-chunk sums
    __shared__ float s_dot1[NN];
    __shared__ float s_dot2[CC];
    __shared__ float s_cnt[CC];
    __shared__ int   s_chunk[NN];
    __shared__ int   s_pos[NN];

    const int b    = blockIdx.x;
    const int tid  = threadIdx.x;
    const int wave = tid >> 5;
    const int lane = tid & 31;

    // ---- stage per-batch indices in LDS ----
    if (tid < NN) {
        s_chunk[tid] = chunk_ids[b * NN + tid];
        s_pos[tid]   = cls_pos[b * NN + tid];
    }
    __syncthreads();

    // ---- counts per chunk (cheap, 16 threads) ----
    if (tid < CC) {
        int cnt = 0;
        #pragma unroll 4
        for (int n = 0; n < NN; ++n) cnt += (s_chunk[n] == tid) ? 1 : 0;
        s_cnt[tid] = (float)cnt;
    }

    // ---- WMMA segment-sum: sums[16,768] = onehot[16,128] @ sent[128,768] ----
    // A (16x4 f32): lanes 0-15 -> M rows, K = {0,1} in v0/v1; lanes 16-31 -> K = {2,3}
    // B (4x16 f32): lane&15 -> column, v0/v1 -> K pair (same K split as A)
    // D (16x16 f32, 8 VGPRs): vgpr r -> chunk r (lanes 0-15) / chunk r+8 (lanes 16-31)
    const int mrow = lane & 15;                 // A: chunk row / B,D: column in tile
    const int kx0  = (lane < 16) ? 0 : 2;       // K index for .x
    const int kx1  = kx0 + 1;                   // K index for .y
    const size_t batch_base = (size_t)b * TT * HH;

    for (int t = wave; t < HTILES; t += NWAVE) {   // 6 tiles per wave, uniform per wave
        const int h0 = t * 16;
        v8f acc = {};
        #pragma unroll 4
        for (int k0 = 0; k0 < NN; k0 += 4) {
            const int c0 = s_chunk[k0 + kx0];
            const int c1 = s_chunk[k0 + kx1];
            v2f a;
            a.x = (c0 == mrow) ? 1.0f : 0.0f;
            a.y = (c1 == mrow) ? 1.0f : 0.0f;
            const size_t r0 = batch_base + (size_t)s_pos[k0 + kx0] * HH + h0 + mrow;
            const size_t r1 = batch_base + (size_t)s_pos[k0 + kx1] * HH + h0 + mrow;
            v2f bm;
            bm.x = hidden[r0];
            bm.y = hidden[r1];
            // D = A * B + C, full f32 (V_WMMA_F32_16X16X4_F32)
            acc = __builtin_amdgcn_wmma_f32_16x16x4_f32(
                /*neg_a=*/false, a, /*neg_b=*/false, bm,
                /*c_mod=*/(short)0, acc, /*reuse_a=*/false, /*reuse_b=*/false);
        }
        // scatter D tile into LDS (conflict-free: unique (chunk, column) per store)
        const int chi = (lane < 16) ? 0 : 8;
        #pragma unroll
        for (int r = 0; r < 8; ++r) {
            s_sums[r + chi][h0 + mrow] = acc[r];
        }
    }
    __syncthreads();

    // ---- dot1[n] = dot(sent_n, W[0:H]) : 16 sentences per wave ----
    for (int i = 0; i < NN / NWAVE; ++i) {
        const int n = wave * (NN / NWAVE) + i;
        const float* row = hidden + batch_base + (size_t)s_pos[n] * HH;
        float p = 0.0f;
        #pragma unroll 6
        for (int j = lane; j < HH; j += 32) p += row[j] * W[j];
        p = wave_reduce_add(p);
        if (lane == 0) s_dot1[n] = p;
    }

    // ---- dot2[c] = dot(sums[c]/max(cnt,1), W[H:2H]) : 2 chunks per wave ----
    for (int i = 0; i < CC / NWAVE; ++i) {
        const int c = wave + i * NWAVE;
        float p = 0.0f;
        #pragma unroll 6
        for (int j = lane; j < HH; j += 32) p += s_sums[c][j] * W[HH + j];
        p = wave_reduce_add(p);
        if (lane == 0) s_dot2[c] = p / fmaxf(s_cnt[c], 1.0f);
    }
    __syncthreads();

    // ---- final logits ----
    if (tid < NN) {
        out[b * NN + tid] = s_dot1[tid] + s_dot2[s_chunk[tid]] + bias[0];
    }
}

extern "C" void kernel_launch(void* const* d_in, const int* in_sizes, int n_in,
                              void* d_out, int out_size, void* d_ws, size_t ws_size,
                              hipStream_t stream) {
    const float* hidden    = (const float*)d_in[0];   // [B,T,H]
    const float* W         = (const float*)d_in[1];   // [1,2H]
    const float* bias      = (const float*)d_in[2];   // [1]
    const int*   cls_pos   = (const int*)d_in[3];     // [B,N]
    const int*   chunk_ids = (const int*)d_in[4];     // [B,N]
    float*       out       = (float*)d_out;           // [B,N]

    hibert_kernel<<<BB, BDIM, 0, stream>>>(hidden, W, bias, cls_pos, chunk_ids, out);
}